// ConvSTFT_82626580840475
// MI455X (gfx1250) — compile-verified
//
#include <hip/hip_runtime.h>

typedef __attribute__((ext_vector_type(2))) float v2f;
typedef __attribute__((ext_vector_type(4))) float v4f;
typedef __attribute__((ext_vector_type(8))) float v8f;

#define T_LEN   160000
#define HOP     320
#define NFFT    1024
#define PAD     512
#define FRAMES  501
#define KROWS   1026
#define NBINS   513
#define LDS_STRIDE 516   // 512-sample chunk + 4 pad dwords -> conflict-free ds_load_b64
#define CHUNK   512

__global__ __launch_bounds__(256)
void conv_stft_wmma(const float* __restrict__ x,
                    const float* __restrict__ basis,
                    float* __restrict__ out) {
  __shared__ float lds[16 * LDS_STRIDE];   // 33 KB

  const int tid    = threadIdx.x;
  const int lane   = tid & 31;
  const int wave   = tid >> 5;
  const int laneM  = lane & 15;    // M index for A, N index for B/D
  const int half   = lane >> 4;    // selects K pair {0,1} vs {2,3}
  const int colOff = half * 2;

  const int f0    = blockIdx.y * 16;      // frame-tile base
  const int b     = blockIdx.z;           // batch
  const int mbase = blockIdx.x * 256;     // 8 waves x 2 tiles x 16 rows
  const int m0_0  = mbase + (wave * 2 + 0) * 16;
  const int m0_1  = mbase + (wave * 2 + 1) * 16;
  const bool m_active = (m0_0 < KROWS);   // wave-uniform -> EXEC all-ones in WMMA region

  // clamp A rows so padded tiles read valid memory (results discarded at store)
  const int rowA0 = (m0_0 + laneM < KROWS) ? (m0_0 + laneM) : (KROWS - 1);
  const int rowA1 = (m0_1 + laneM < KROWS) ? (m0_1 + laneM) : (KROWS - 1);

  const float* xb = x + (size_t)b * T_LEN;

  v8f acc0 = {};
  v8f acc1 = {};

  for (int chunk = 0; chunk < 2; ++chunk) {
    const int cbase = chunk * CHUNK;

    // ---- stage 16 frame rows x 512 samples into LDS, zero-padding OOB ----
    // pad boundaries are float4-aligned: (512 - 320*f) % 4 == 0 for all f
    #pragma unroll
    for (int i = 0; i < 8; ++i) {
      const int flat = tid + i * 256;       // 0..2047 float4 slots
      const int row  = flat >> 7;           // frame row 0..15 (128 float4 per row)
      const int col  = (flat & 127) << 2;   // 0..508
      const int g    = (f0 + row) * HOP + cbase + col - PAD;
      v4f v = {};
      if (g >= 0 && g <= T_LEN - 4) v = *(const v4f*)(xb + g);
      *(v4f*)(&lds[row * LDS_STRIDE + col]) = v;
    }
    __syncthreads();

    if (m_active) {
      const float* pA0 = basis + (size_t)rowA0 * NFFT + cbase + colOff;
      const float* pA1 = basis + (size_t)rowA1 * NFFT + cbase + colOff;
      const float* pB  = &lds[laneM * LDS_STRIDE + colOff];
      #pragma unroll 4
      for (int n0 = 0; n0 < CHUNK; n0 += 4) {
        const v2f a0 = *(const v2f*)(pA0 + n0);   // A: 16x4 f32 fragment
        const v2f a1 = *(const v2f*)(pA1 + n0);
        const v2f bf = *(const v2f*)(pB + n0);    // B: 4x16 f32 fragment (LDS)
        acc0 = __builtin_amdgcn_wmma_f32_16x16x4_f32(false, a0, false, bf,
                                                     (short)0, acc0, false, false);
        acc1 = __builtin_amdgcn_wmma_f32_16x16x4_f32(false, a1, false, bf,
                                                     (short)0, acc1, false, false);
      }
    }
    __syncthreads();
  }

  // ---- scatter D tiles: row m -> (real bin m) or (imag bin m-513) ----
  if (m_active) {
    const int f = f0 + laneM;               // N = lane & 15
    if (f < FRAMES) {
      #pragma unroll
      for (int j = 0; j < 8; ++j) {
        const int M = j + half * 8;         // C/D layout: lanes 16-31 hold M+8
        int m = m0_0 + M;
        if (m < KROWS) {
          const int ri = (m < NBINS) ? 0 : 1;
          const int k  = (m < NBINS) ? m : (m - NBINS);
          out[(((size_t)b * NBINS + k) * FRAMES + f) * 2 + ri] = acc0[j];
        }
        m = m0_1 + M;
        if (m < KROWS) {
          const int ri = (m < NBINS) ? 0 : 1;
          const int k  = (m < NBINS) ? m : (m - NBINS);
          out[(((size_t)b * NBINS + k) * FRAMES + f) * 2 + ri] = acc1[j];
        }
      }
    }
  }
}

extern "C" void kernel_launch(void* const* d_in, const int* in_sizes, int n_in,
                              void* d_out, int out_size, void* d_ws, size_t ws_size,
                              hipStream_t stream) {
  (void)in_sizes; (void)n_in; (void)out_size; (void)d_ws; (void)ws_size;
  const float* x     = (const float*)d_in[0];   // (32, 160000) f32
  const float* basis = (const float*)d_in[1];   // (1026, 1024) f32
  float* out = (float*)d_out;                   // (32, 513, 501, 2) f32

  dim3 grid(5, 32, 32);   // 5 m-blocks (256 rows each), 32 frame-tiles, 32 batches
  conv_stft_wmma<<<grid, 256, 0, stream>>>(x, basis, out);
}